// PostProcess_6493990552135
// MI455X (gfx1250) — compile-verified
//
#include <hip/hip_runtime.h>
#include <hip/hip_bf16.h>
#include <stdint.h>

// ---------------- problem constants ----------------
#define BB    64
#define QQ    4096
#define CC    256
#define QC    (QQ * CC)          // 1,048,576 scores per batch
#define TOPK  100

#define NBINS        4096        // top-12 float bits (order-preserving for x>=0)
#define BLK_PER_B    32
#define SEG          (QC / BLK_PER_B)   // 32768 elements per block
#define CHUNK        4096        // floats per TDM tile (16 KB)
#define NCHUNK       (SEG / CHUNK)      // 8
#define CAND_MAX     2048

#define HIST_BYTES   (BB * NBINS * 4)   // 1 MB

typedef unsigned int u32x4 __attribute__((ext_vector_type(4)));
typedef int          i32x8 __attribute__((ext_vector_type(8)));
typedef int          i32x4 __attribute__((ext_vector_type(4)));

// fused score; must be bit-identical between hist and collect passes
static __device__ __forceinline__ float fused_score(float logit, float act) {
    return act / (1.0f + __expf(-logit));
}

static __device__ __forceinline__ uint32_t lds_off(const void* p) {
    // generic shared pointer: low 32 bits == LDS byte offset
    return (uint32_t)(uintptr_t)p;
}

// ---- Tensor Data Mover: 1-D contiguous f32 tile, global -> LDS ----
static __device__ __forceinline__ void tdm_load_f32(uint32_t lds_addr,
                                                    const float* gptr,
                                                    uint32_t nelem) {
    uint64_t ga = (uint64_t)(uintptr_t)gptr;
    u32x4 g0;
    g0[0] = 1u;                                    // count=1, user mode, no gather
    g0[1] = lds_addr;                              // LDS byte address
    g0[2] = (uint32_t)(ga & 0xFFFFFFFFu);          // global addr [31:0]
    g0[3] = (uint32_t)((ga >> 32) & 0x01FFFFFFu)   // global addr [56:32]
          | (2u << 30);                            // type = 2 ("image")
    i32x8 g1;
    g1[0] = (int)(2u << 16);                       // data_size = 4 bytes
    g1[1] = (int)((nelem & 0xFFFFu) << 16);        // tensor_dim0 [15:0]
    g1[2] = (int)((nelem >> 16) & 0xFFFFu)         // tensor_dim0 [31:16]
          | (int)(1u << 16);                       // tensor_dim1 = 1 (low 16)
    g1[3] = (int)((nelem & 0xFFFFu) << 16);        // tile_dim0 = nelem
    g1[4] = 1;                                     // tile_dim1 = 1, tile_dim2 = 0
    g1[5] = (int)nelem;                            // tensor_dim0_stride (low 32)
    g1[6] = 0;
    g1[7] = 0;
    i32x4 z4 = {0, 0, 0, 0};                       // dims 2..4 unused (2-D tile)
    i32x8 z8 = {0, 0, 0, 0, 0, 0, 0, 0};
    // clang-23 / therock-10.0 form: 6 args (g0, g1, g2, g3, extra, cpol)
    __builtin_amdgcn_tensor_load_to_lds(g0, g1, z4, z4, z8, 0);
}

// ================= Kernel 1: fused score + histogram (TDM-streamed) =================
__global__ __launch_bounds__(256)
void pp_hist_kernel(const float* __restrict__ logits,
                    const float* __restrict__ act,
                    unsigned*    __restrict__ ghist) {
    __shared__ float    buf[2][CHUNK];     // 32 KB double buffer
    __shared__ unsigned lhist[NBINS];      // 16 KB

    const int b   = blockIdx.y;
    const int blk = blockIdx.x;
    const int t   = threadIdx.x;
    const bool wave0 = (t >> 5) == 0;

    for (int i = t; i < NBINS; i += 256) lhist[i] = 0u;

    const float* gbase = logits + (size_t)b * QC + (size_t)blk * SEG;

    if (wave0) {
        tdm_load_f32(lds_off(&buf[0][0]), gbase, CHUNK);
    }
    __syncthreads();   // histogram zero visible; pipeline primed

    for (int c = 0; c < NCHUNK; ++c) {
        if (wave0) {
            if (c + 1 < NCHUNK) {
                tdm_load_f32(lds_off(&buf[(c + 1) & 1][0]),
                             gbase + (size_t)(c + 1) * CHUNK, CHUNK);
                __builtin_amdgcn_s_wait_tensorcnt(1);  // chunk c complete
            } else {
                __builtin_amdgcn_s_wait_tensorcnt(0);
            }
        }
        __syncthreads();                   // chunk c visible to all waves

        const float* sb = &buf[c & 1][0];
        const int qbase = blk * (SEG / CC) + c * (CHUNK / CC);
        #pragma unroll
        for (int i = 0; i < CHUNK / 256; ++i) {
            // all 256 lanes share one query -> scalar (s_load) actionness
            float a = act[b * QQ + qbase + i];
            float x = sb[i * 256 + t];
            float p = fused_score(x, a);
            unsigned key = __float_as_uint(p) >> 20;   // 12-bit monotone key
            atomicAdd(&lhist[key], 1u);
        }
        __syncthreads();                   // buffer free for reuse
    }

    for (int i = t; i < NBINS; i += 256) {
        unsigned v = lhist[i];
        if (v) atomicAdd(&ghist[b * NBINS + i], v);
    }
}

// ================= Kernel 2: per-batch threshold bin =================
__global__ __launch_bounds__(256)
void pp_thresh_kernel(const unsigned* __restrict__ ghist,
                      unsigned*       __restrict__ tkey) {
    __shared__ unsigned h[NBINS];
    __shared__ unsigned ps[256];
    const int b = blockIdx.x;
    const int t = threadIdx.x;

    for (int i = t; i < NBINS; i += 256) h[i] = ghist[b * NBINS + i];
    __syncthreads();

    unsigned s = 0;
    #pragma unroll
    for (int i = 0; i < 16; ++i) s += h[t * 16 + i];
    ps[t] = s;
    __syncthreads();

    if (t == 0) {
        unsigned acc = 0;
        int T = 0;
        for (int j = 255; j >= 0; --j) {
            if (acc + ps[j] >= (unsigned)TOPK) {
                for (int bin = j * 16 + 15; bin >= j * 16; --bin) {
                    acc += h[bin];
                    if (acc >= (unsigned)TOPK) { T = bin; break; }
                }
                break;
            }
            acc += ps[j];
        }
        tkey[b] = (unsigned)T;
    }
}

// ================= Kernel 3: collect candidates >= threshold =================
__global__ __launch_bounds__(256)
void pp_collect_kernel(const float* __restrict__ logits,
                       const float* __restrict__ act,
                       const unsigned* __restrict__ tkey,
                       unsigned* __restrict__ ccount,
                       unsigned long long* __restrict__ cand) {
    const int b   = blockIdx.y;
    const int blk = blockIdx.x;
    const int t   = threadIdx.x;
    const unsigned T = tkey[b];

    const float4* lp = (const float4*)(logits + (size_t)b * QC + (size_t)blk * SEG);

    #pragma unroll 4
    for (int i = 0; i < SEG / (256 * 4); ++i) {
        const int e4 = i * 256 + t;          // float4 index inside segment
        const int e0 = e4 * 4;               // element offset inside segment
        float4 x = lp[e4];
        const int q = (blk * SEG + e0) >> 8; // 4 elems share one query (4 | 256)
        float a = act[b * QQ + q];
        float p[4] = { fused_score(x.x, a), fused_score(x.y, a),
                       fused_score(x.z, a), fused_score(x.w, a) };
        #pragma unroll
        for (int k = 0; k < 4; ++k) {
            unsigned bits = __float_as_uint(p[k]);
            if ((bits >> 20) >= T) {
                unsigned pos = atomicAdd(&ccount[b], 1u);
                if (pos < CAND_MAX) {
                    unsigned flat = (unsigned)(blk * SEG + e0 + k);
                    cand[b * CAND_MAX + pos] =
                        ((unsigned long long)bits << 32) | (0xFFFFFFFFu - flat);
                }
            }
        }
    }
}

// ================= Kernel 4: bitonic sort + emit outputs =================
__global__ __launch_bounds__(1024)
void pp_emit_kernel(const unsigned long long* __restrict__ cand,
                    const unsigned* __restrict__ ccount,
                    const float* __restrict__ segs,
                    const float* __restrict__ tsz,
                    float* __restrict__ out) {
    __shared__ unsigned long long keys[CAND_MAX];   // 16 KB
    const int b = blockIdx.x;
    const int t = threadIdx.x;

    unsigned n = ccount[b];
    if (n > CAND_MAX) n = CAND_MAX;

    for (int j = t; j < CAND_MAX; j += 1024)
        keys[j] = (j < (int)n) ? cand[b * CAND_MAX + j] : 0ull;
    __syncthreads();

    // bitonic sort, descending (score desc, index asc on ties)
    for (unsigned k = 2; k <= CAND_MAX; k <<= 1) {
        for (unsigned j = k >> 1; j > 0; j >>= 1) {
            unsigned idx = (t & (j - 1)) | ((t & ~(j - 1)) << 1);
            unsigned ixj = idx | j;
            unsigned long long A = keys[idx];
            unsigned long long Bv = keys[ixj];
            bool desc = ((idx & k) == 0);
            if (desc ? (A < Bv) : (A > Bv)) { keys[idx] = Bv; keys[ixj] = A; }
            __syncthreads();
        }
    }

    if (t < TOPK) {
        unsigned long long key = keys[t];
        bool valid = (t < (int)n);
        unsigned bits = (unsigned)(key >> 32);
        unsigned flat = 0xFFFFFFFFu - (unsigned)(key & 0xFFFFFFFFu);
        float score = valid ? __uint_as_float(bits) : 0.0f;
        unsigned lab = valid ? (flat & (CC - 1)) : 0u;
        unsigned q   = valid ? (flat >> 8) : 0u;

        float cen = segs[((size_t)b * QQ + q) * 2 + 0];
        float wid = segs[((size_t)b * QQ + q) * 2 + 1];
        float dur = tsz[b];
        float t1 = valid ? (cen - 0.5f * wid) * dur : 0.0f;
        float t2 = valid ? (cen + 0.5f * wid) * dur : 0.0f;

        const int o = b * TOPK + t;
        out[o] = score;                                      // scores   [B,100]
        out[BB * TOPK + o] = (float)lab;                     // labels   [B,100]
        out[2 * BB * TOPK + o * 2 + 0] = t1;                 // segments [B,100,2]
        out[2 * BB * TOPK + o * 2 + 1] = t2;
        out[4 * BB * TOPK + o] = (float)q;                   // query_ids[B,100]
    }
}

// ================= host launcher =================
extern "C" void kernel_launch(void* const* d_in, const int* in_sizes, int n_in,
                              void* d_out, int out_size, void* d_ws, size_t ws_size,
                              hipStream_t stream) {
    const float* logits = (const float*)d_in[0];   // [B,Q,C]
    const float* act    = (const float*)d_in[1];   // [B,Q,1]
    const float* segs   = (const float*)d_in[2];   // [B,Q,2]
    const float* tsz    = (const float*)d_in[3];   // [B]
    float* out = (float*)d_out;

    char* ws = (char*)d_ws;
    unsigned* ghist  = (unsigned*)ws;                               // 1 MB
    unsigned* ccount = (unsigned*)(ws + HIST_BYTES);                // 256 B
    unsigned* tkey   = (unsigned*)(ws + HIST_BYTES + 256);          // 256 B
    unsigned long long* cand =
        (unsigned long long*)(ws + HIST_BYTES + 512);               // 1 MB

    (void)hipMemsetAsync(ws, 0, HIST_BYTES + 512, stream);

    pp_hist_kernel   <<<dim3(BLK_PER_B, BB), 256, 0, stream>>>(logits, act, ghist);
    pp_thresh_kernel <<<BB, 256, 0, stream>>>(ghist, tkey);
    pp_collect_kernel<<<dim3(BLK_PER_B, BB), 256, 0, stream>>>(logits, act, tkey,
                                                               ccount, cand);
    pp_emit_kernel   <<<BB, 1024, 0, stream>>>(cand, ccount, segs, tsz, out);
}